// ChebyshevConvolutionLin_20590073217364
// MI455X (gfx1250) — compile-verified
//
#include <hip/hip_runtime.h>

#define DH 64   // hidden / input feature dim

typedef __attribute__((ext_vector_type(2))) float v2f;
typedef __attribute__((ext_vector_type(8))) float v8f;

// ---------------------------------------------------------------------------
// deg[i] = number of edges whose source (row) == i
// ---------------------------------------------------------------------------
__global__ void deg_count_kernel(const long long* __restrict__ row,
                                 float* __restrict__ deg, int nE) {
    int e = blockIdx.x * blockDim.x + threadIdx.x;
    if (e < nE) atomicAdd(&deg[(int)row[e]], 1.0f);
}

// dis[i] = deg>0 ? rsqrt(max(deg,1)) : 0
__global__ void dis_kernel(const float* __restrict__ deg,
                           float* __restrict__ dis, int n) {
    int i = blockIdx.x * blockDim.x + threadIdx.x;
    if (i < n) {
        float d = deg[i];
        dis[i] = (d > 0.0f) ? rsqrtf(fmaxf(d, 1.0f)) : 0.0f;
    }
}

// int64 -> int32 indices, per-edge weight norm = -(dis[row]*dis[col])
__global__ void edge_prep_kernel(const long long* __restrict__ ei,
                                 const float* __restrict__ dis,
                                 int* __restrict__ row32, int* __restrict__ col32,
                                 float* __restrict__ norm, int nE) {
    int e = blockIdx.x * blockDim.x + threadIdx.x;
    if (e < nE) {
        int r = (int)ei[e];
        int c = (int)ei[nE + e];
        row32[e] = r;
        col32[e] = c;
        norm[e]  = -(dis[r] * dis[c]);
    }
}

// ---------------------------------------------------------------------------
// prop: out[col_e, :] += norm_e * z[row_e, :]   (out pre-zeroed)
// 16 lanes per edge, float4 gather, 4 f32 atomics per lane.
// All arrays fit in the 192MB L2 -> gathers/atomics are L2-resident.
// ---------------------------------------------------------------------------
__global__ __launch_bounds__(256)
void prop_scatter_kernel(const float* __restrict__ z,
                         const float* __restrict__ norm,
                         const int* __restrict__ row,
                         const int* __restrict__ col,
                         float* __restrict__ out, unsigned int nWork) {
    unsigned int gid = blockIdx.x * blockDim.x + threadIdx.x;
    if (gid >= nWork) return;
    unsigned int e  = gid >> 4;
    unsigned int f4 = (gid & 15u) << 2;
    float w = norm[e];
    if (w == 0.0f) return;                 // isolated endpoint -> contributes 0
    int r = row[e];
    int c = col[e];
    float4 zv = *(const float4*)(z + (size_t)r * DH + f4);
    float* o = out + (size_t)c * DH + f4;
    atomicAdd(o + 0, w * zv.x);
    atomicAdd(o + 1, w * zv.y);
    atomicAdd(o + 2, w * zv.z);
    atomicAdd(o + 3, w * zv.w);
}

// t2 = 2*t2 - t0  (Chebyshev recurrence), float4 elementwise
__global__ void cheb_combine_kernel(float4* __restrict__ t2,
                                    const float4* __restrict__ t0,
                                    unsigned int n4) {
    unsigned int i = blockIdx.x * blockDim.x + threadIdx.x;
    if (i < n4) {
        float4 a = t2[i];
        float4 b = t0[i];
        a.x = 2.0f * a.x - b.x;
        a.y = 2.0f * a.y - b.y;
        a.z = 2.0f * a.z - b.z;
        a.w = 2.0f * a.w - b.w;
        t2[i] = a;
    }
}

// ---------------------------------------------------------------------------
// Fused WMMA GEMM: out[n,64] = relu( T0@W[0] + T1@W[1] + T2@W[2] + bias )
// f32 V_WMMA_F32_16X16X4_F32 to preserve exact fp32 reference semantics.
// One wave per 16x64 output strip: 4 accumulator tiles, 3*16 K-steps.
//
// A frag (16x4 f32, 2 VGPRs/lane): lanes 0-15 hold K={k0,k0+1} of row M=lane,
//                                  lanes 16-31 hold K={k0+2,k0+3} of row M=lane-16.
// B frag (4x16 f32, 2 VGPRs/lane): lanes 0-15 hold rows K={k0,k0+1} at N=lane,
//                                  lanes 16-31 hold rows K={k0+2,k0+3} at N=lane-16.
// C/D   (16x16 f32, 8 VGPRs): VGPR v -> row v (lanes 0-15) / row v+8 (lanes 16-31).
// ---------------------------------------------------------------------------
__global__ __launch_bounds__(256)
void cheb_gemm_bias_relu_kernel(const float* __restrict__ T0,
                                const float* __restrict__ T1,
                                const float* __restrict__ T2,
                                const float* __restrict__ W,     // [3,64,64]
                                const float* __restrict__ bias,  // [64]
                                float* __restrict__ out,         // [n,64]
                                int nNodes, int doRelu) {
    const int lane  = threadIdx.x & 31;
    const int wave  = threadIdx.x >> 5;
    const int mBase = blockIdx.x * 128 + wave * 16;
    if (mBase >= nNodes) return;          // uniform per wave: EXEC stays all-ones

    const int l15  = lane & 15;
    const int half = lane >> 4;           // 0 or 1
    const int kOff = half * 2;

    int aRow = mBase + l15;
    if (aRow > nNodes - 1) aRow = nNodes - 1;   // clamp tail rows (discarded on store)

    const float* Asrc[3] = {T0, T1, T2};
    v8f acc[4] = {v8f{0}, v8f{0}, v8f{0}, v8f{0}};

#pragma unroll
    for (int c = 0; c < 3; ++c) {
        const float* A  = Asrc[c] + (size_t)aRow * DH + kOff;
        const float* Wc = W + c * DH * DH;
#pragma unroll
        for (int k0 = 0; k0 < DH; k0 += 4) {
            v2f a;
            a.x = A[k0 + 0];
            a.y = A[k0 + 1];
#pragma unroll
            for (int nt = 0; nt < 4; ++nt) {
                const int ncol = nt * 16 + l15;
                v2f b;
                b.x = Wc[(k0 + kOff + 0) * DH + ncol];
                b.y = Wc[(k0 + kOff + 1) * DH + ncol];
                acc[nt] = __builtin_amdgcn_wmma_f32_16x16x4_f32(
                    false, a, false, b, (short)0, acc[nt], false, false);
            }
        }
    }

    float bv[4];
#pragma unroll
    for (int nt = 0; nt < 4; ++nt) bv[nt] = bias[nt * 16 + l15];

#pragma unroll
    for (int v = 0; v < 8; ++v) {
        const int r = mBase + v + half * 8;
        if (r < nNodes) {
#pragma unroll
            for (int nt = 0; nt < 4; ++nt) {
                float val = acc[nt][v] + bv[nt];
                if (doRelu) val = fmaxf(val, 0.0f);
                out[(size_t)r * DH + nt * 16 + l15] = val;
            }
        }
    }
}

// logits[n,2] = h[n,:] @ Wlin + blin
__global__ void final_linear_kernel(const float* __restrict__ h,
                                    const float* __restrict__ Wlin,   // [64,2]
                                    const float* __restrict__ blin,   // [2]
                                    float* __restrict__ out, int nNodes) {
    int n = blockIdx.x * blockDim.x + threadIdx.x;
    if (n >= nNodes) return;
    float a0 = blin[0], a1 = blin[1];
    const float* hp = h + (size_t)n * DH;
#pragma unroll 8
    for (int f = 0; f < DH; ++f) {
        float v = hp[f];
        a0 = fmaf(v, Wlin[f * 2 + 0], a0);
        a1 = fmaf(v, Wlin[f * 2 + 1], a1);
    }
    out[(size_t)n * 2 + 0] = a0;
    out[(size_t)n * 2 + 1] = a1;
}

// ---------------------------------------------------------------------------
extern "C" void kernel_launch(void* const* d_in, const int* in_sizes, int n_in,
                              void* d_out, int out_size, void* d_ws, size_t ws_size,
                              hipStream_t stream) {
    const float*     x    = (const float*)d_in[0];
    const long long* ei   = (const long long*)d_in[1];
    const float*     W1   = (const float*)d_in[2];
    const float*     b1   = (const float*)d_in[3];
    const float*     W2   = (const float*)d_in[4];
    const float*     b2   = (const float*)d_in[5];
    const float*     Wlin = (const float*)d_in[6];
    const float*     blin = (const float*)d_in[7];

    const int nNodes = in_sizes[0] / DH;
    const int nE     = in_sizes[1] / 2;

    // ---- carve workspace ----
    char* p = (char*)d_ws;
    auto carve = [&](size_t bytes) -> char* {
        char* q = p;
        p += (bytes + 255) & ~(size_t)255;
        return q;
    };
    const size_t featBytes = (size_t)nNodes * DH * sizeof(float);
    float* deg   = (float*)carve((size_t)nNodes * sizeof(float));
    float* dis   = (float*)carve((size_t)nNodes * sizeof(float));
    int*   row32 = (int*)  carve((size_t)nE * sizeof(int));
    int*   col32 = (int*)  carve((size_t)nE * sizeof(int));
    float* norm  = (float*)carve((size_t)nE * sizeof(float));
    float* T1    = (float*)carve(featBytes);
    float* T2    = (float*)carve(featBytes);
    float* H1    = (float*)carve(featBytes);
    float* H2    = (float*)carve(featBytes);

    const int TB = 256;
    const int gE = (nE + TB - 1) / TB;
    const int gN = (nNodes + TB - 1) / TB;
    const unsigned int nWork = (unsigned int)nE * 16u;
    const int gProp = (int)((nWork + TB - 1) / TB);
    const unsigned int n4 = (unsigned int)nNodes * (DH / 4);
    const int gC = (int)((n4 + TB - 1) / TB);
    const int gGemm = (nNodes + 127) / 128;

    // ---- graph normalization (shared by both layers) ----
    hipMemsetAsync(deg, 0, (size_t)nNodes * sizeof(float), stream);
    deg_count_kernel<<<gE, TB, 0, stream>>>(ei, deg, nE);
    dis_kernel<<<gN, TB, 0, stream>>>(deg, dis, nNodes);
    edge_prep_kernel<<<gE, TB, 0, stream>>>(ei, dis, row32, col32, norm, nE);

    // ---- one ChebConv layer ----
    auto layer = [&](const float* in, const float* W, const float* b, float* out, int relu) {
        hipMemsetAsync(T1, 0, featBytes, stream);
        prop_scatter_kernel<<<gProp, TB, 0, stream>>>(in, norm, row32, col32, T1, nWork);
        hipMemsetAsync(T2, 0, featBytes, stream);
        prop_scatter_kernel<<<gProp, TB, 0, stream>>>(T1, norm, row32, col32, T2, nWork);
        cheb_combine_kernel<<<gC, TB, 0, stream>>>((float4*)T2, (const float4*)in, n4);
        cheb_gemm_bias_relu_kernel<<<gGemm, TB, 0, stream>>>(in, T1, T2, W, b, out, nNodes, relu);
    };

    layer(x, W1, b1, H1, 1);
    layer(H1, W2, b2, H2, 1);

    final_linear_kernel<<<gN, TB, 0, stream>>>(H2, Wlin, blin, (float*)d_out, nNodes);

    // second tuple element: edge_index (int64) raw bytes behind the logits
    hipMemcpyAsync((char*)d_out + (size_t)nNodes * 2 * sizeof(float),
                   (const void*)ei, (size_t)nE * 2 * sizeof(long long),
                   hipMemcpyDeviceToDevice, stream);
}